// SelectConv_88338887344631
// MI455X (gfx1250) — compile-verified
//
#include <hip/hip_runtime.h>
#include <cstdint>

// Problem constants (fixed by the reference: imgs (8,32,256,256) f32,
// kernel (32,1,5,5) f32, stride=1, pad=2, dil=1, groups=32 -> depthwise).
#define IMG_H 256
#define IMG_W 256
#define NCH   32
#define NB    8
#define KS    5
#define HALO  2

// Tile: 64x16 outputs per block of 256 threads (8 wave32s).
#define TX    64
#define TY    16
#define IN_W  (TX + 2 * HALO)   // 68
#define IN_H  (TY + 2 * HALO)   // 20
#define LDSW  69                // odd row stride -> bank-conflict-free row pairs
#define NTHREADS 256

__global__ __launch_bounds__(NTHREADS)
void selectconv_maxplus_dw5x5(const float* __restrict__ imgs,
                              const float* __restrict__ krn,
                              float* __restrict__ out)
{
    __shared__ float tile[IN_H * LDSW];

    const int tid = threadIdx.x;
    const int gx0 = blockIdx.x * TX;
    const int gy0 = blockIdx.y * TY;
    const int bc  = blockIdx.z;                 // b * NCH + c
    const int c   = bc & (NCH - 1);

    const float* __restrict__ plane = imgs + (size_t)bc * (IMG_H * IMG_W);
    const float NEG = -__builtin_inff();

    // Border blocks: pre-fill the LDS tile with the max-plus neutral (-inf)
    // so EXEC-masked (OOB) async lanes leave correct padding behind.
    const bool edge = (gx0 == 0) || (gy0 == 0) ||
                      (gx0 + TX == IMG_W) || (gy0 + TY == IMG_H);
    if (edge) {
        for (int i = tid; i < IN_H * LDSW; i += NTHREADS) tile[i] = NEG;
        __syncthreads();
    }

    // ---- Async halo-tile copy: global -> LDS via the CDNA5 async data path.
    // Each lane issues GLOBAL_LOAD_ASYNC_TO_LDS_B32 with its own global
    // address (VGPR pair) and its own LDS byte address (VGPR). Tracked by
    // ASYNCcnt, not LOADcnt.
    const uint32_t lds_base = (uint32_t)(uintptr_t)(&tile[0]); // low 32 bits of
                                                               // flat LDS ptr = LDS offset
    #pragma unroll
    for (int k = 0; k < (IN_H * IN_W + NTHREADS - 1) / NTHREADS; ++k) {
        const int i = tid + k * NTHREADS;
        if (i < IN_H * IN_W) {
            const int iy = i / IN_W;
            const int ix = i - iy * IN_W;
            const int gy = gy0 - HALO + iy;
            const int gx = gx0 - HALO + ix;
            if ((unsigned)gy < (unsigned)IMG_H && (unsigned)gx < (unsigned)IMG_W) {
                const uint64_t ga = (uint64_t)(uintptr_t)(plane + gy * IMG_W + gx);
                const uint32_t la = lds_base + (uint32_t)(iy * LDSW + ix) * 4u;
                asm volatile("global_load_async_to_lds_b32 %0, %1, off"
                             :: "v"(la), "v"(ga) : "memory");
            }
        }
    }
    // Wave waits for its own async transfers; barrier makes all waves' data visible.
    asm volatile("s_wait_asynccnt 0" ::: "memory");
    __syncthreads();

    // ---- Kernel taps: channel is block-uniform -> scalar-cached loads.
    float kw[KS * KS];
    const float* __restrict__ kp = krn + c * (KS * KS);
    #pragma unroll
    for (int i = 0; i < KS * KS; ++i) kw[i] = kp[i];

    // ---- Compute: each thread produces 4 consecutive x outputs.
    const int tx = (tid & 15) * 4;   // output x within tile
    const int ty = tid >> 4;         // output y within tile

    float a0 = NEG, a1 = NEG, a2 = NEG, a3 = NEG;

    #pragma unroll
    for (int ky = 0; ky < KS; ++ky) {
        float r[8];
        const int rb = (ty + ky) * LDSW + tx;
        #pragma unroll
        for (int j = 0; j < 8; ++j) r[j] = tile[rb + j];
        #pragma unroll
        for (int kx = 0; kx < KS; ++kx) {
            const float kv = kw[ky * KS + kx];
            a0 = fmaxf(a0, r[kx + 0] + kv);
            a1 = fmaxf(a1, r[kx + 1] + kv);
            a2 = fmaxf(a2, r[kx + 2] + kv);
            a3 = fmaxf(a3, r[kx + 3] + kv);
        }
    }

    // ---- Coalesced 16B store per thread.
    float* __restrict__ op = out + (size_t)bc * (IMG_H * IMG_W)
                                 + (size_t)(gy0 + ty) * IMG_W + (gx0 + tx);
    *(float4*)op = make_float4(a0, a1, a2, a3);
}

extern "C" void kernel_launch(void* const* d_in, const int* in_sizes, int n_in,
                              void* d_out, int out_size, void* d_ws, size_t ws_size,
                              hipStream_t stream) {
    const float* imgs = (const float*)d_in[0];   // (8,32,256,256) f32
    const float* krn  = (const float*)d_in[1];   // (32,1,5,5) f32
    float* out = (float*)d_out;                  // (8,32,256,256) f32

    dim3 grid(IMG_W / TX, IMG_H / TY, NB * NCH); // 4 x 16 x 256 = 16384 blocks
    dim3 block(NTHREADS);
    selectconv_maxplus_dw5x5<<<grid, block, 0, stream>>>(imgs, krn, out);
}